// GAT_Block_8959301779839
// MI455X (gfx1250) — compile-verified
//
#include <hip/hip_runtime.h>

typedef float v2f __attribute__((ext_vector_type(2)));
typedef float v8f __attribute__((ext_vector_type(8)));

#define Bk   32
#define Ck   3
#define Fk   2048
#define Nk   25
#define Hk   64
#define FN   (Fk * Nk)      /* 51200 rows per batch */
#define TILE 400            /* rows per workgroup = 16 groups of 25 = 25 subtiles of 16 */

__global__ __launch_bounds__(256) void gat_kernel(
    const float* __restrict__ x,    // [B, C, F, N]
    const float* __restrict__ mask, // [N, N]
    const float* __restrict__ W,    // [C, H]
    const float* __restrict__ bW,   // [H]
    const float* __restrict__ a2,   // [H]
    float* __restrict__ out)        // [B, H, F, N]
{
    __shared__ float s2s[TILE];
    __shared__ float diag[TILE];
    __shared__ float md[Nk];
    __shared__ float w2s[4];        // W@a2 (3) and bW.a2 (1)

    const int tid = threadIdx.x;

    // ---- Phase 0: tiny precomputes ----
    if (tid < Nk) md[tid] = mask[tid * Nk + tid];
    if (tid >= 32 && tid < 35) {            // w2[c] = sum_h W[c,h]*a2[h]
        int c = tid - 32;
        float s = 0.f;
        for (int h = 0; h < Hk; ++h) s += W[c * Hk + h] * a2[h];
        w2s[c] = s;
    }
    if (tid == 35) {                        // b2 = bW . a2
        float s = 0.f;
        for (int h = 0; h < Hk; ++h) s += bW[h] * a2[h];
        w2s[3] = s;
    }
    __syncthreads();

    const int rowbase = blockIdx.x * TILE;  // tile never crosses a batch (51200 % 400 == 0)
    const int b       = rowbase / FN;
    const int fnbase  = rowbase % FN;
    const float* xb   = x + b * (Ck * FN);

    // ---- Phase A: s2 per row (3-term dot, coalesced) ----
    for (int r = tid; r < TILE; r += 256) {
        int fn = fnbase + r;
        float x0 = xb[fn];
        float x1 = xb[FN + fn];
        float x2 = xb[2 * FN + fn];
        s2s[r] = x0 * w2s[0] + x1 * w2s[1] + x2 * w2s[2] + w2s[3];
    }
    __syncthreads();

    // ---- Phase B: softmax over each 25-node group, times mask diagonal ----
    if (tid < 16) {
        int g = tid * Nk;
        float m = s2s[g];
        for (int i = 1; i < Nk; ++i) m = fmaxf(m, s2s[g + i]);
        float s = 0.f;
        for (int i = 0; i < Nk; ++i) s += __expf(s2s[g + i] - m);
        float inv = 1.0f / s;
        for (int i = 0; i < Nk; ++i)
            diag[g + i] = __expf(s2s[g + i] - m) * inv * md[i];
    }
    __syncthreads();

    // ---- Phase C: h = x@W + bW via V_WMMA_F32_16X16X4_F32, scale by diag, store ----
    const int  lane = tid & 31;
    const int  wave = tid >> 5;
    const int  lo   = lane & 15;
    const bool low  = lane < 16;

    // A operand: W^T chunk [16h x 4c]; lane = M = h, VGPR0/1 = K {0,1} (low half) / {2,3} (high half).
    // Loop-invariant: preload all four 16-wide H chunks.
    v2f aW[4];
    v8f cB[4];
    for (int hc = 0; hc < 4; ++hc) {
        int h = hc * 16 + lo;
        aW[hc].x = low ? W[0 * Hk + h] : W[2 * Hk + h];
        aW[hc].y = low ? W[1 * Hk + h] : 0.0f;       // padded K=3 row is zero
        int hb = hc * 16 + (low ? 0 : 8);            // C/D layout: VGPR v -> M = v (+8 for hi lanes)
        for (int v = 0; v < 8; ++v) cB[hc][v] = bW[hb + v];
    }

    for (int t = wave; t < TILE / 16; t += 8) {      // 25 subtiles of 16 rows, uniform per wave
        int fn0 = fnbase + t * 16;
        int fn  = fn0 + lo;

        // B operand: X [4c x 16fn]; lane = N = fn, VGPR0/1 = K {0,1} (low) / {2,3} (high).
        float xc1 = xb[FN + fn];
        v2f bx;
        bx.x = low ? xb[fn] : xb[2 * FN + fn];
        bx.y = low ? xc1 : 0.0f;

        float dscale = diag[t * 16 + lo];            // per-column (fn) scale, broadcast over rows

        for (int hc = 0; hc < 4; ++hc) {
            v8f d = __builtin_amdgcn_wmma_f32_16x16x4_f32(
                false, aW[hc], false, bx, (short)0, cB[hc], false, false);
            // D: lane -> fn = fn0+lo, VGPR v -> h = hc*16 + v (+8 for hi lanes)
            float* p = out + b * (Hk * FN) + (hc * 16 + (low ? 0 : 8)) * FN + fn0 + lo;
            #pragma unroll
            for (int v = 0; v < 8; ++v)
                p[v * FN] = d[v] * dscale;           // 2x contiguous 64B segments per store
        }
    }
}

extern "C" void kernel_launch(void* const* d_in, const int* in_sizes, int n_in,
                              void* d_out, int out_size, void* d_ws, size_t ws_size,
                              hipStream_t stream) {
    const float* x    = (const float*)d_in[0];
    const float* mask = (const float*)d_in[1];
    const float* W    = (const float*)d_in[2];
    const float* bW   = (const float*)d_in[3];
    // d_in[4] = a1, d_in[6] = ab: cancel exactly inside the row softmax (constant along axis j)
    const float* a2   = (const float*)d_in[5];
    float* out        = (float*)d_out;

    const int rows   = Bk * FN;          // 1,638,400
    const int blocks = rows / TILE;      // 4096
    hipLaunchKernelGGL(gat_kernel, dim3(blocks), dim3(256), 0, stream,
                       x, mask, W, bW, a2, out);
}